// ConditionalDLFactorized18_74680891343540
// MI455X (gfx1250) — compile-verified
//
#include <hip/hip_runtime.h>
#include <math.h>

// ---------------- problem constants ----------------
static constexpr int TT = 2048, BBATCH = 8, CC = 512, OO = 512, NEXP = 16;
static constexpr int TB = TT * BBATCH;           // 16384 tokens
static constexpr int NSLOT = 2 * TB + 16 * 15;   // 33008 (max padded assignment slots)
static constexpr int NTILE = NSLOT / 16;         // 2063 expert-GEMM tiles
static constexpr int DUMMY = 2 * TB;             // first dummy inter row (pad slots land here)
static constexpr size_t INTER_ROWS = (size_t)DUMMY + 16;

// ---------------- workspace layout (bytes) — total ~64.7 MB ----------------
static constexpr size_t alignup(size_t x) { return (x + 255) & ~(size_t)255; }
static constexpr size_t WS_INTER = 0;                                              // inter [INTER_ROWS, OO] f32
static constexpr size_t WS_PART  = alignup(WS_INTER + INTER_ROWS * OO * 4);        // partials [TB/16, 16] f32
static constexpr size_t WS_RE    = alignup(WS_PART + (size_t)(TB / 16) * NEXP * 4);// route_e [2*TB] int
static constexpr size_t WS_RG    = alignup(WS_RE + (size_t)2 * TB * 4);            // route_g [2*TB] f32
static constexpr size_t WS_TOK   = alignup(WS_RG + (size_t)2 * TB * 4);            // tok_list [NSLOT] int
static constexpr size_t WS_DST   = alignup(WS_TOK + (size_t)NSLOT * 4);            // dest_list [NSLOT] int
static constexpr size_t WS_TEXP  = alignup(WS_DST + (size_t)NSLOT * 4);            // tile_expert [NTILE] int
static constexpr size_t WS_CNT   = alignup(WS_TEXP + (size_t)NTILE * 4);           // counts [16] int
static constexpr size_t WS_OFF   = alignup(WS_CNT + 64);                           // offsets [17] int
static constexpr size_t WS_CUR   = alignup(WS_OFF + 128);                          // cursor [16] int

typedef float v2f __attribute__((ext_vector_type(2)));
typedef float v8f __attribute__((ext_vector_type(8)));

__device__ __forceinline__ v8f wmma_f32_k4(v2f a, v2f b, v8f c) {
  return __builtin_amdgcn_wmma_f32_16x16x4_f32(false, a, false, b, (short)0, c,
                                               false, false);
}

// ---------------- kernel 0: zero expert counts ----------------
__global__ void k_zero(int* __restrict__ counts) {
  if (threadIdx.x < NEXP) counts[threadIdx.x] = 0;
}

// ---------------- kernel 1: WMMA gating + top-2 softmax + routing decisions ----
__global__ __launch_bounds__(32) void k_gate(
    const float* __restrict__ x, const float* __restrict__ aw,
    const float* __restrict__ ab, float* __restrict__ partials,
    int* __restrict__ route_e, float* __restrict__ route_g,
    int* __restrict__ counts) {
  __shared__ float lE[16 * 17];
  __shared__ float lR[16 * 17];
  const int tile = blockIdx.x;  // 16 tokens per tile
  const int lane = threadIdx.x;
  const int half = lane >> 4;
  const int l15 = lane & 15;
  const int tok0 = tile * 16;

  v8f acc = (v8f){0.f, 0.f, 0.f, 0.f, 0.f, 0.f, 0.f, 0.f};
  const float* xrow = x + (size_t)(tok0 + l15) * CC;  // A row M = l15
  const float* wrow = aw + (size_t)l15 * CC;          // B col N = l15
  for (int k = 0; k < CC; k += 4) {
    const int kk = k + half * 2;  // lanes 0-15: K=k,k+1 ; lanes 16-31: K=k+2,k+3
    v2f a = {xrow[kk], xrow[kk + 1]};
    v2f b = {wrow[kk], wrow[kk + 1]};
    acc = wmma_f32_k4(a, b, acc);
  }
#pragma unroll
  for (int v = 0; v < 8; ++v) lE[(v + 8 * half) * 17 + l15] = acc[v];
  __syncthreads();

  if (lane < 16) {
    const int m = lane;
    float vals[NEXP];
    float v1 = -3.4e38f, v2 = -3.4e38f;
    int i1 = 0, i2 = 0;
    for (int j = 0; j < NEXP; ++j) {
      const float v = lE[m * 17 + j] + ab[j];
      vals[j] = v;
      if (v > v1) { v2 = v1; i2 = i1; v1 = v; i1 = j; }
      else if (v > v2) { v2 = v; i2 = j; }
    }
    float s = 0.f, r[NEXP];
    for (int j = 0; j < NEXP; ++j) {
      const float mv = (vals[j] < v2) ? -1.0e6f : vals[j];  // keeptopk_masked
      const float e = __expf(mv - v1);                      // underflows to 0 for masked
      r[j] = e; s += e;
    }
    const float inv = 1.f / s;
    for (int j = 0; j < NEXP; ++j) { r[j] *= inv; lR[m * 17 + j] = r[j]; }
    const int n = tok0 + m;
    route_e[2 * n + 0] = i1; route_e[2 * n + 1] = i2;
    route_g[2 * n + 0] = r[i1]; route_g[2 * n + 1] = r[i2];
    atomicAdd(&counts[i1], 1);
    atomicAdd(&counts[i2], 1);
  }
  __syncthreads();
  if (lane < 16) {  // fixed-order per-tile importance partials (deterministic loss)
    float s = 0.f;
    for (int m = 0; m < 16; ++m) s += lR[m * 17 + lane];
    partials[tile * NEXP + lane] = s;
  }
}

// ---------------- kernel 2: segment offsets (padded to 16) ----------------
__global__ void k_offsets(const int* __restrict__ counts, int* __restrict__ offsets,
                          int* __restrict__ cursor) {
  if (threadIdx.x == 0) {
    int off = 0;
    for (int e = 0; e < NEXP; ++e) {
      offsets[e] = off;
      off += (counts[e] + 15) & ~15;
    }
    offsets[NEXP] = off;
  }
  if (threadIdx.x < NEXP) cursor[threadIdx.x] = 0;
}

// ---------------- kernel 3: default-fill slots + tile->expert map ----------
__global__ void k_fill(const int* __restrict__ offsets, int* __restrict__ tok_list,
                       int* __restrict__ dest_list, int* __restrict__ tile_expert) {
  const int s = blockIdx.x * blockDim.x + threadIdx.x;
  if (s >= NSLOT) return;
  tok_list[s] = 0;                    // pad slots read token 0 (in-range)
  dest_list[s] = DUMMY + (s & 15);    // pad results go to write-only dummy rows
  if ((s & 15) == 0) {
    int e = 0;
    if (s < offsets[NEXP]) {
      for (int j = 0; j < NEXP; ++j)
        if (s >= offsets[j]) e = j;
    }
    tile_expert[s >> 4] = e;
  }
}

// ---------------- kernel 4: slot placement (int atomics only) --------------
__global__ void k_place(const int* __restrict__ route_e, const int* __restrict__ offsets,
                        int* __restrict__ cursor, int* __restrict__ tok_list,
                        int* __restrict__ dest_list) {
  const int n = blockIdx.x * blockDim.x + threadIdx.x;
  if (n >= TB) return;
  for (int r = 0; r < 2; ++r) {
    const int e = route_e[2 * n + r];
    const int pos = atomicAdd(&cursor[e], 1);
    const int s = offsets[e] + pos;
    tok_list[s] = n;
    dest_list[s] = 2 * n + r;  // result row depends only on (token, rank) -> deterministic y
  }
}

// ---------------- kernel 5: dynamic-bias GEMM, writes y = x@bias_w.T + b ---
__global__ __launch_bounds__(256) void k_bias(const float* __restrict__ x,
                                              const float* __restrict__ bw,
                                              const float* __restrict__ bb,
                                              float* __restrict__ y) {
  __shared__ float lX[16 * 516];  // 516-f32 stride: conflict-free across 16 rows
  const int t = blockIdx.x;
  const int tid = threadIdx.x;
  {  // coalesced stage of 16 token rows
    const int r = tid >> 4, c4 = tid & 15;
    const float4* src = (const float4*)(x + (size_t)(t * 16 + r) * CC);
    float* dst = lX + r * 516;
#pragma unroll
    for (int i = 0; i < 8; ++i) {
      const int cq = c4 + i * 16;
      *(float4*)(dst + cq * 4) = src[cq];
    }
  }
  __syncthreads();
  const int wv = tid >> 5, lane = tid & 31, half = lane >> 4, l15 = lane & 15;
  v8f acc[4];
#pragma unroll
  for (int j = 0; j < 4; ++j) acc[j] = (v8f){0.f, 0.f, 0.f, 0.f, 0.f, 0.f, 0.f, 0.f};
  const float* arow = lX + l15 * 516;
  for (int k = 0; k < CC; k += 4) {
    const int kk = k + half * 2;
    v2f a = {arow[kk], arow[kk + 1]};
#pragma unroll
    for (int j = 0; j < 4; ++j) {
      const int n = wv * 64 + j * 16 + l15;
      const float* wr = bw + (size_t)n * CC;
      v2f b = {wr[kk], wr[kk + 1]};
      acc[j] = wmma_f32_k4(a, b, acc[j]);
    }
  }
#pragma unroll
  for (int j = 0; j < 4; ++j) {
    const int n = wv * 64 + j * 16 + l15;
    const float bias = bb[n];
#pragma unroll
    for (int v = 0; v < 8; ++v) {
      const int m = v + 8 * half;
      y[(size_t)(t * 16 + m) * OO + n] = acc[j][v] + bias;
    }
  }
}

// ---------------- kernel 6: grouped expert GEMM -> inter rows --------------
__global__ __launch_bounds__(256) void k_expert(
    const float* __restrict__ x, const float* __restrict__ w1,
    const int* __restrict__ tok_list, const int* __restrict__ dest_list,
    const int* __restrict__ tile_expert, float* __restrict__ inter) {
  __shared__ float lX[16 * 516];
  __shared__ int ltok[16];
  __shared__ int ldst[16];
  const int t = blockIdx.x;
  const int tid = threadIdx.x;
  if (tid < 16) {
    ltok[tid] = tok_list[t * 16 + tid];
    ldst[tid] = dest_list[t * 16 + tid];
  }
  __syncthreads();
  {  // gather 16 routed token rows into LDS
    const int r = tid >> 4, c4 = tid & 15;
    const float4* src = (const float4*)(x + (size_t)ltok[r] * CC);
    float* dst = lX + r * 516;
#pragma unroll
    for (int i = 0; i < 8; ++i) {
      const int cq = c4 + i * 16;
      *(float4*)(dst + cq * 4) = src[cq];
    }
  }
  __syncthreads();
  const int e = tile_expert[t];
  const int wv = tid >> 5, lane = tid & 31, half = lane >> 4, l15 = lane & 15;
  const float* W = w1 + (size_t)e * OO * CC;  // [OUT, C] for this expert
  v8f acc[4];
#pragma unroll
  for (int j = 0; j < 4; ++j) acc[j] = (v8f){0.f, 0.f, 0.f, 0.f, 0.f, 0.f, 0.f, 0.f};
  const float* arow = lX + l15 * 516;
  for (int k = 0; k < CC; k += 4) {
    const int kk = k + half * 2;
    v2f a = {arow[kk], arow[kk + 1]};
#pragma unroll
    for (int j = 0; j < 4; ++j) {
      const int n = wv * 64 + j * 16 + l15;
      const float* wr = W + (size_t)n * CC;
      v2f b = {wr[kk], wr[kk + 1]};
      acc[j] = wmma_f32_k4(a, b, acc[j]);
    }
  }
#pragma unroll
  for (int j = 0; j < 4; ++j) {
    const int n = wv * 64 + j * 16 + l15;
#pragma unroll
    for (int v = 0; v < 8; ++v) {
      const int m = v + 8 * half;
      inter[(size_t)ldst[m] * OO + n] = acc[j][v];
    }
  }
}

// ---------------- kernel 7: combine y += g0*i0 + g1*i1 (fixed order) -------
__global__ void k_combine(const float* __restrict__ inter,
                          const float* __restrict__ route_g, float* __restrict__ y) {
  const size_t idx = (size_t)blockIdx.x * blockDim.x + threadIdx.x;
  if (idx >= (size_t)TB * OO) return;
  const size_t n = idx / OO;
  const size_t o = idx - n * OO;
  const float g0 = route_g[2 * n + 0];
  const float g1 = route_g[2 * n + 1];
  y[idx] += g0 * inter[(2 * n + 0) * (size_t)OO + o] +
            g1 * inter[(2 * n + 1) * (size_t)OO + o];
}

// ---------------- kernel 8: loss = 0.01 * std(importance, ddof=1)/mean -----
__global__ void k_loss(const float* __restrict__ partials, float* __restrict__ loss_out) {
  __shared__ float imp[NEXP];
  const int lane = threadIdx.x;
  if (lane < NEXP) {
    float s = 0.f;
    for (int b = 0; b < TB / 16; ++b) s += partials[b * NEXP + lane];
    imp[lane] = s;
  }
  __syncthreads();
  if (lane == 0) {
    float mean = 0.f;
    for (int j = 0; j < NEXP; ++j) mean += imp[j];
    mean /= (float)NEXP;
    float var = 0.f;
    for (int j = 0; j < NEXP; ++j) {
      const float d = imp[j] - mean;
      var += d * d;
    }
    var /= (float)(NEXP - 1);  // unbiased, like torch.std
    loss_out[0] = 0.01f * sqrtf(var) / mean;
  }
}

extern "C" void kernel_launch(void* const* d_in, const int* in_sizes, int n_in,
                              void* d_out, int out_size, void* d_ws, size_t ws_size,
                              hipStream_t stream) {
  (void)in_sizes; (void)n_in; (void)out_size; (void)ws_size;
  const float* x  = (const float*)d_in[0];
  const float* aw = (const float*)d_in[1];
  const float* ab = (const float*)d_in[2];
  const float* w1 = (const float*)d_in[3];
  const float* bw = (const float*)d_in[4];
  const float* bb = (const float*)d_in[5];
  float* y = (float*)d_out;
  float* loss = y + (size_t)TB * OO;

  char* ws = (char*)d_ws;
  float* inter   = (float*)(ws + WS_INTER);
  float* part    = (float*)(ws + WS_PART);
  int*   re      = (int*)(ws + WS_RE);
  float* rg      = (float*)(ws + WS_RG);
  int*   tokl    = (int*)(ws + WS_TOK);
  int*   dstl    = (int*)(ws + WS_DST);
  int*   texp    = (int*)(ws + WS_TEXP);
  int*   counts  = (int*)(ws + WS_CNT);
  int*   offs    = (int*)(ws + WS_OFF);
  int*   cur     = (int*)(ws + WS_CUR);

  k_zero<<<1, 32, 0, stream>>>(counts);
  k_gate<<<TB / 16, 32, 0, stream>>>(x, aw, ab, part, re, rg, counts);
  k_offsets<<<1, 32, 0, stream>>>(counts, offs, cur);
  k_fill<<<(NSLOT + 255) / 256, 256, 0, stream>>>(offs, tokl, dstl, texp);
  k_place<<<TB / 256, 256, 0, stream>>>(re, offs, cur, tokl, dstl);
  k_bias<<<TB / 16, 256, 0, stream>>>(x, bw, bb, y);
  k_expert<<<NTILE, 256, 0, stream>>>(x, w1, tokl, dstl, texp, inter);
  k_combine<<<(int)(((size_t)TB * OO + 255) / 256), 256, 0, stream>>>(inter, rg, y);
  k_loss<<<1, 32, 0, stream>>>(part, loss);
}